// ChainCRF_40948218200488
// MI455X (gfx1250) — compile-verified
//
#include <hip/hip_runtime.h>
#include <hip/hip_bf16.h>
#include <math.h>

typedef float v2f __attribute__((ext_vector_type(2)));
typedef float v8f __attribute__((ext_vector_type(8)));
typedef unsigned int u32x4 __attribute__((ext_vector_type(4)));
typedef int i32x8 __attribute__((ext_vector_type(8)));
typedef int i32x4 __attribute__((ext_vector_type(4)));

#define L1C   49
#define NCOL  2401          // 49*49
#define NPAD  2432          // padded to multiple of 64
#define KTOT  256
#define TLEN  512
#define BATCH 32
#define MTOT  (BATCH * TLEN) // 16384

#define BM 64
#define BN 64
#define BK 32
#define LDA 36               // LDS row stride in dwords (conflict-free, 16B aligned)

// ---------------------------------------------------------------------------
// Kernel 0: fold W_s into W_t:  Wc[i][n] = W_t[i][n] + W_s[i][n % 49]
// ---------------------------------------------------------------------------
__global__ void crf_prep_w(const float* __restrict__ Wt,
                           const float* __restrict__ Ws,
                           float* __restrict__ Wc) {
  int idx = blockIdx.x * blockDim.x + threadIdx.x;
  const int total = KTOT * NPAD;
  if (idx >= total) return;
  int i = idx / NPAD;
  int n = idx - i * NPAD;
  float v = 0.f;
  if (n < NCOL) v = Wt[i * NCOL + n] + Ws[i * L1C + (n % L1C)];
  Wc[idx] = v;
}

// ---------------------------------------------------------------------------
// Kernel 1: energy GEMM with fp32 WMMA (V_WMMA_F32_16X16X4_F32).
// E[m][n] = (sum_k X[m][k] * Wc[k][n] + bias[n]) * mask[m],  m = b*512+t.
// ---------------------------------------------------------------------------
__global__ __launch_bounds__(128)
void crf_energy_gemm(const float* __restrict__ X,    // [MTOT][KTOT]
                     const float* __restrict__ Wc,   // [KTOT][NPAD]
                     const float* __restrict__ bias, // [NCOL]
                     const float* __restrict__ mask, // [MTOT]
                     float* __restrict__ E) {        // [MTOT][NPAD]
  __shared__ float As[BM * LDA];       // A tile, [row][k]
  __shared__ float Bs[BN * LDA];       // W tile transposed, [n][k]

  const int tid  = threadIdx.x;
  const int wave = tid >> 5;
  const int lane = tid & 31;
  const int lo   = lane & 15;
  const int hi   = lane >> 4;

  const int m0 = blockIdx.x * BM;
  const int n0 = blockIdx.y * BN;

  v8f acc[4];
  v8f zero = {};
  #pragma unroll
  for (int t = 0; t < 4; ++t) acc[t] = zero;

  for (int kc = 0; kc < KTOT; kc += BK) {
    #pragma unroll
    for (int it = 0; it < 4; ++it) {
      const int row = it * 16 + (tid >> 3);
      const int kk  = (tid & 7) * 4;
      const float4 v = *(const float4*)(X + (size_t)(m0 + row) * KTOT + kc + kk);
      *(float4*)(&As[row * LDA + kk]) = v;
    }
    {
      const int k  = tid >> 2;         // 0..31
      const int ng = (tid & 3) * 16;   // 0,16,32,48
      #pragma unroll
      for (int q = 0; q < 4; ++q) {
        const int n = ng + q * 4;
        const float4 v = *(const float4*)(Wc + (size_t)(kc + k) * NPAD + n0 + n);
        Bs[(n + 0) * LDA + k] = v.x;
        Bs[(n + 1) * LDA + k] = v.y;
        Bs[(n + 2) * LDA + k] = v.z;
        Bs[(n + 3) * LDA + k] = v.w;
      }
    }
    __syncthreads();

    if (kc + BK < KTOT) {
      const int row = tid >> 3;
      __builtin_prefetch(X + (size_t)(m0 + row) * KTOT + kc + BK, 0, 1);
    }

    #pragma unroll
    for (int ks = 0; ks < BK; ks += 4) {
      const v2f a = *(const v2f*)(&As[(wave * 16 + lo) * LDA + ks + 2 * hi]);
      #pragma unroll
      for (int t = 0; t < 4; ++t) {
        const v2f bb = *(const v2f*)(&Bs[(t * 16 + lo) * LDA + ks + 2 * hi]);
        acc[t] = __builtin_amdgcn_wmma_f32_16x16x4_f32(
            false, a, false, bb, (short)0, acc[t], false, false);
      }
    }
    __syncthreads();
  }

  #pragma unroll
  for (int t = 0; t < 4; ++t) {
    const int gn = n0 + t * 16 + lo;
    if (gn < NCOL) {
      const float bv = bias[gn];
      #pragma unroll
      for (int vr = 0; vr < 8; ++vr) {
        const int gm = m0 + wave * 16 + hi * 8 + vr;
        E[(size_t)gm * NPAD + gn] = (acc[t][vr] + bv) * mask[gm];
      }
    }
  }
}

// ---------------------------------------------------------------------------
// TDM helper: DMA one padded energy row (2432 fp32, 9728 B) into LDS.
// D# group0: count=1 | lds_addr | global_addr[56:0] | type=2
// D# group1: data_size=4B, tensor_dim0=tile_dim0=2432, stride0=2432
// This toolchain's builtin takes 6 args (extra i32x8 group before cpol).
// ---------------------------------------------------------------------------
#if __has_builtin(__builtin_amdgcn_tensor_load_to_lds)
#define HAVE_TDM 1
__device__ __forceinline__ void tdm_load_row(const float* g, unsigned lds_off) {
  const unsigned long long ga = (unsigned long long)g;
  u32x4 g0;
  g0[0] = 1u;                                            // count=1, user mode
  g0[1] = lds_off;                                       // lds_addr (bytes)
  g0[2] = (unsigned)(ga & 0xFFFFFFFFull);                // global_addr[31:0]
  g0[3] = (unsigned)((ga >> 32) & 0x01FFFFFFull)         // global_addr[56:32]
          | (2u << 30);                                  // type=2 ("image")
  i32x8 g1;
  g1[0] = 0x20000;                                       // data_size=2 (4 bytes)
  g1[1] = (int)((NPAD & 0xFFFF) << 16);                  // tensor_dim0[15:0]
  g1[2] = 0;                                             // tensor_dim0 hi, dim1 lo
  g1[3] = (int)(NPAD << 16);                             // tile_dim0
  g1[4] = 0;                                             // tile_dim1/2 unused
  g1[5] = NPAD;                                          // tensor_dim0_stride lo
  g1[6] = 0;
  g1[7] = 0;
  i32x4 z4 = {0, 0, 0, 0};
  i32x8 z8 = {0, 0, 0, 0, 0, 0, 0, 0};
  __builtin_amdgcn_tensor_load_to_lds(g0, g1, z4, z4, z8, 0);
}
#else
#define HAVE_TDM 0
#endif

// ---------------------------------------------------------------------------
// Kernel 2: CRF forward scan. One workgroup (64 thr) per batch element.
// Double-buffered energy tiles in LDS, prefetched by the Tensor Data Mover
// while the 49-way logsumexp recurrence computes the current step.
// ---------------------------------------------------------------------------
__global__ __launch_bounds__(64)
void crf_scan(const float* __restrict__ E,       // [MTOT][NPAD]
              const int*   __restrict__ target,  // [BATCH][TLEN]
              const float* __restrict__ mask,    // [MTOT]
              float* __restrict__ out) {         // [BATCH]
  __shared__ float etile[2][NPAD];
  __shared__ float pvec[L1C];
  __shared__ float nvec[L1C];
  __shared__ float red[64];
  __shared__ float tgtE_sh;

  const int b   = blockIdx.x;
  const int tid = threadIdx.x;
  const size_t base = (size_t)b * TLEN;

  // ---- gold path energy: sum_t E[b,t, prev(t), target(t)] ----
  float acc = 0.f;
  for (int t = tid; t < TLEN; t += 64) {
    const int tg = target[base + t];
    const int pv = (t == 0) ? (L1C - 1) : target[base + t - 1];
    acc += E[(base + t) * (size_t)NPAD + pv * L1C + tg];
  }
  red[tid] = acc;
  __syncthreads();
  for (int s = 32; s > 0; s >>= 1) {
    if (tid < s) red[tid] += red[tid + s];
    __syncthreads();
  }
  if (tid == 0) tgtE_sh = red[0];
  __syncthreads();

  // ---- p0 = energy[b, 0, 48, :] ----
  if (tid < L1C) pvec[tid] = E[base * (size_t)NPAD + (L1C - 1) * L1C + tid];

#if HAVE_TDM
  // prologue: TDM-prefetch tile for t=1 into buffer 1
  tdm_load_row(E + (base + 1) * (size_t)NPAD,
               (unsigned)(unsigned long long)(void*)&etile[1][0]);
#endif
  __syncthreads();

  // ---- sequential forward recurrence over t = 1..511 ----
  for (int t = 1; t < TLEN; ++t) {
    const size_t m   = base + t;
    const int    cur = t & 1;

#if HAVE_TDM
    if (t + 1 < TLEN) {
      // prefetch next tile while we compute this one
      tdm_load_row(E + (m + 1) * (size_t)NPAD,
                   (unsigned)(unsigned long long)(void*)&etile[(t + 1) & 1][0]);
      __builtin_amdgcn_s_wait_tensorcnt(1);   // tile t landed (in-order)
    } else {
      __builtin_amdgcn_s_wait_tensorcnt(0);
    }
#else
    for (int i = tid; i < NPAD; i += 64) etile[cur][i] = E[m * NPAD + i];
#endif
    __syncthreads();

    if (tid < L1C) {
      float tmp[L1C];
      float mx = -INFINITY;
      #pragma unroll
      for (int j = 0; j < L1C; ++j) {
        const float v = etile[cur][j * L1C + tid] + pvec[j];
        tmp[j] = v;
        mx = fmaxf(mx, v);
      }
      float s = 0.f;
      #pragma unroll
      for (int j = 0; j < L1C; ++j) s += __expf(tmp[j] - mx);
      const float nk = mx + __logf(s);
      const float mt = mask[m];
      nvec[tid] = mt * nk + (1.f - mt) * pvec[tid];
    }
    __syncthreads();
    if (tid < L1C) pvec[tid] = nvec[tid];
    __syncthreads();   // all reads of etile[cur] done before TDM reuses it
  }

  // ---- out[b] = logsumexp_k(p[k]) - tgtE ----
  red[tid] = (tid < L1C) ? pvec[tid] : -INFINITY;
  __syncthreads();
  for (int s = 32; s > 0; s >>= 1) {
    if (tid < s) red[tid] = fmaxf(red[tid], red[tid + s]);
    __syncthreads();
  }
  const float mx = red[0];
  __syncthreads();
  red[tid] = (tid < L1C) ? __expf(pvec[tid] - mx) : 0.f;
  __syncthreads();
  for (int s = 32; s > 0; s >>= 1) {
    if (tid < s) red[tid] += red[tid + s];
    __syncthreads();
  }
  if (tid == 0) out[b] = mx + __logf(red[0]) - tgtE_sh;
}

// ---------------------------------------------------------------------------
// Launcher. d_in order: x, target, mask, W_t, W_s, b.
// Workspace: E fp32 [16384][2432] (159.4 MB) | Wc fp32 [256][2432].
// ---------------------------------------------------------------------------
extern "C" void kernel_launch(void* const* d_in, const int* in_sizes, int n_in,
                              void* d_out, int out_size, void* d_ws, size_t ws_size,
                              hipStream_t stream) {
  const float* x      = (const float*)d_in[0];
  const int*   target = (const int*)  d_in[1];
  const float* mask   = (const float*)d_in[2];
  const float* W_t    = (const float*)d_in[3];
  const float* W_s    = (const float*)d_in[4];
  const float* bias   = (const float*)d_in[5];
  float* out = (float*)d_out;

  float* E  = (float*)d_ws;                         // [MTOT][NPAD]
  float* Wc = E + (size_t)MTOT * NPAD;              // [KTOT][NPAD]

  {
    const int total = KTOT * NPAD;
    crf_prep_w<<<(total + 255) / 256, 256, 0, stream>>>(W_t, W_s, Wc);
  }
  {
    dim3 grid(MTOT / BM, NPAD / BN);                // 256 x 38
    crf_energy_gemm<<<grid, 128, 0, stream>>>(x, Wc, bias, mask, E);
  }
  crf_scan<<<BATCH, 64, 0, stream>>>(E, target, mask, out);
}